// ATACSeqEmbedding_34875134444135
// MI455X (gfx1250) — compile-verified
//
#include <hip/hip_runtime.h>

#define NUM_CHR 24
#define N_EMB   512
#define DIM     512
#define BATCH   1024
#define TILE_M  16
#define MB      2                         // M-tiles per block (B-reuse factor)
#define ROWS    (TILE_M * MB)             // 32 samples per block
#define MAX_PAIRS (BATCH / ROWS)          // 32

typedef __attribute__((ext_vector_type(2))) float v2f;
typedef __attribute__((ext_vector_type(4))) float v4f;
typedef __attribute__((ext_vector_type(8))) float v8f;
typedef __attribute__((ext_vector_type(4))) int   v4i;

#if defined(__has_builtin)
#  if __has_builtin(__builtin_amdgcn_global_load_async_to_lds_b128)
#    define HAVE_ASYNC_LDS 1
#  endif
#endif
#ifndef HAVE_ASYNC_LDS
#  define HAVE_ASYNC_LDS 0
#endif

// ---------------------------------------------------------------------------
// Kernel 0: zero the per-chromosome counters (workspace is poisoned 0xAA).
// ---------------------------------------------------------------------------
__global__ void zero_counts_kernel(int* __restrict__ counts) {
    if (threadIdx.x < NUM_CHR) counts[threadIdx.x] = 0;
}

// ---------------------------------------------------------------------------
// Kernel 1: one wave32 per sample.
//   - compute normalized RBF weights w[b, 0..511] into workspace
//   - bucket sample ids by chromosome (atomicAdd counter + index list)
// ---------------------------------------------------------------------------
__global__ __launch_bounds__(256) void weights_kernel(
    const int*   __restrict__ chromosome,   // [B]
    const float* __restrict__ position,     // [B,1]
    const float* __restrict__ centers,      // [NUM_CHR][N_EMB]
    const float* __restrict__ logvar,       // [NUM_CHR][N_EMB]
    float* __restrict__ w_out,              // [B][N_EMB]
    int*   __restrict__ counts,             // [NUM_CHR]
    int*   __restrict__ bucket)             // [NUM_CHR][BATCH]
{
    const int wave = threadIdx.x >> 5;
    const int lane = threadIdx.x & 31;
    const int b = blockIdx.x * 8 + wave;
    if (b >= BATCH) return;

    const int   c = chromosome[b];
    const float p = position[b];
    const float* __restrict__ cen = centers + c * N_EMB;
    const float* __restrict__ lv  = logvar  + c * N_EMB;

    float vals[16];
    float s = 0.0f;
    #pragma unroll
    for (int j = 0; j < 16; ++j) {
        const int n = lane + 32 * j;
        const float d   = p - cen[n];
        const float var = __expf(lv[n]);
        const float e   = __expf(-(d * d) / (2.0f * var));
        vals[j] = e;
        s += e;
    }
    // wave32 butterfly reduction
    #pragma unroll
    for (int off = 16; off >= 1; off >>= 1)
        s += __shfl_xor(s, off, 32);
    const float inv = 1.0f / s;

    float* __restrict__ wp = w_out + (size_t)b * N_EMB;
    #pragma unroll
    for (int j = 0; j < 16; ++j)
        wp[lane + 32 * j] = vals[j] * inv;

    if (lane == 0) {
        const int slot = atomicAdd(&counts[c], 1);
        bucket[c * BATCH + slot] = b;
    }
}

// ---------------------------------------------------------------------------
// Kernel 2: per-chromosome dense GEMM via V_WMMA_F32_16X16X4_F32.
//   block = 256 threads (8 waves). Grid: (32-sample pair, chromosome, d/128).
//   Each wave owns a 16-wide D slice and TWO 16-row accumulators that share
//   every B fetch:  out[32x16] = W[32x512] @ E_c[512x16].
//   A tiles are staged Global->LDS with the gfx1250 async-copy path and
//   XOR-swizzled so the 16-lane A reads are bank-conflict-free without
//   padding (exactly 64 KB LDS).
// ---------------------------------------------------------------------------
__global__ __launch_bounds__(256) void rbf_gemm_kernel(
    const float* __restrict__ embeddings,   // [NUM_CHR][N_EMB][DIM]
    const float* __restrict__ w,            // [B][N_EMB]
    const int*   __restrict__ counts,       // [NUM_CHR]
    const int*   __restrict__ bucket,       // [NUM_CHR][BATCH]
    float* __restrict__ out)                // [B][DIM]
{
    __shared__ float s_w[ROWS * N_EMB];     // exactly 64 KB, swizzled

    const int c    = blockIdx.y;
    const int pair = blockIdx.x;
    const int cnt  = counts[c];
    if (pair * ROWS >= cnt) return;         // block-uniform: EXEC stays all-1s

    const int base = c * BATCH + pair * ROWS;   // into bucket[]

    // ---- stage 32x512 weight tile into LDS (swizzle: col ^ 4*(row&15)) ----
    for (int i = threadIdx.x; i < ROWS * (N_EMB / 4); i += 256) {
        const int row  = i / (N_EMB / 4);
        const int col4 = i % (N_EMB / 4);
        const int idx  = pair * ROWS + row;
        const int dst  = row * N_EMB + ((col4 * 4) ^ ((row & 15) * 4));
        if (idx < cnt) {
            const int bid = bucket[base + row];
            const float* src = w + (size_t)bid * N_EMB + col4 * 4;
#if HAVE_ASYNC_LDS
            __builtin_amdgcn_global_load_async_to_lds_b128(
                (__attribute__((address_space(1))) v4i*)src,
                (__attribute__((address_space(3))) v4i*)&s_w[dst],
                /*offset=*/0, /*cpol=*/0);
#else
            *(v4f*)&s_w[dst] = *(const v4f*)src;
#endif
        } else {
            const v4f zero4 = {0.f, 0.f, 0.f, 0.f};
            *(v4f*)&s_w[dst] = zero4;       // zero weights -> zero partials
        }
    }
#if HAVE_ASYNC_LDS
#  if __has_builtin(__builtin_amdgcn_s_wait_asynccnt)
    __builtin_amdgcn_s_wait_asynccnt(0);
#  else
    asm volatile("s_wait_asynccnt 0x0" ::: "memory");
#  endif
#endif
    __syncthreads();

    const int wave = threadIdx.x >> 5;
    const int lane = threadIdx.x & 31;
    const int half = lane >> 4;       // selects K pair {0,1} vs {2,3}
    const int lm   = lane & 15;       // A: row M ; B/C: column N
    const int swz  = lm * 4;          // per-row LDS column swizzle
    const int d0   = blockIdx.z * 128 + wave * 16;

    // this lane's output column within E_c
    const float* __restrict__ ecol =
        embeddings + (size_t)c * N_EMB * DIM + d0 + lm;

    v8f acc[MB] = {};
    #pragma unroll 4
    for (int kk = 0; kk < N_EMB; kk += 4) {
        const int kh = kk + 2 * half;           // K index for this lane half
        const int ks = kh ^ swz;                // swizzled LDS column (pair-safe)
        // B (4x16, K x N): lane=N; VGPR0/1 = rows kh, kh+1 of E_c
        v2f bm;
        bm.x = ecol[(size_t)kh * DIM];
        bm.y = ecol[(size_t)(kh + 1) * DIM];
        // A tiles (16x4 each): lane=M; shared B across both accumulators
        v2f a0, a1;
        a0.x = s_w[lm * N_EMB + ks];
        a0.y = s_w[lm * N_EMB + ks + 1];
        a1.x = s_w[(16 + lm) * N_EMB + ks];
        a1.y = s_w[(16 + lm) * N_EMB + ks + 1];
        acc[0] = __builtin_amdgcn_wmma_f32_16x16x4_f32(
            false, a0, false, bm, (short)0, acc[0], false, false);
        acc[1] = __builtin_amdgcn_wmma_f32_16x16x4_f32(
            false, a1, false, bm, (short)0, acc[1], false, false);
    }

    // C/D layout: VGPR r, lanes 0-15 -> M=r ; lanes 16-31 -> M=r+8 ; N=lm
    #pragma unroll
    for (int t = 0; t < MB; ++t) {
        #pragma unroll
        for (int r = 0; r < 8; ++r) {
            const int m   = t * TILE_M + r + 8 * half;
            const int idx = pair * ROWS + m;
            if (idx < cnt) {
                const int bid = bucket[base + m];
                out[(size_t)bid * DIM + d0 + lm] = acc[t][r];
            }
        }
    }
}

// ---------------------------------------------------------------------------
extern "C" void kernel_launch(void* const* d_in, const int* in_sizes, int n_in,
                              void* d_out, int out_size, void* d_ws, size_t ws_size,
                              hipStream_t stream) {
    (void)in_sizes; (void)n_in; (void)out_size; (void)ws_size;

    const int*   chromosome = (const int*)  d_in[0];
    const float* position   = (const float*)d_in[1];
    const float* embeddings = (const float*)d_in[2];
    const float* centers    = (const float*)d_in[3];
    const float* logvar     = (const float*)d_in[4];
    float* out = (float*)d_out;

    // workspace layout: counts (pad 256B) | bucket 96KB | weights 2MB
    char*  ws     = (char*)d_ws;
    int*   counts = (int*)ws;
    int*   bucket = (int*)(ws + 256);
    float* wbuf   = (float*)(ws + 256 + NUM_CHR * BATCH * sizeof(int));

    zero_counts_kernel<<<1, 32, 0, stream>>>(counts);

    weights_kernel<<<BATCH / 8, 256, 0, stream>>>(
        chromosome, position, centers, logvar, wbuf, counts, bucket);

    dim3 grid(MAX_PAIRS, NUM_CHR, DIM / 128);
    rbf_gemm_kernel<<<grid, 256, 0, stream>>>(
        embeddings, wbuf, counts, bucket, out);
}